// LayoutPoint_77867757077174
// MI455X (gfx1250) — compile-verified
//
#include <hip/hip_runtime.h>

// LayoutPoint: out[b,i,j] = max_n relu(1-|i-127*u[b,n]|) * relu(1-|j-127*v[b,n]|)
// W = H = 128, NUM = 256, B = 16.  Tent support < 2px  =>  each point hits a
// 2x2 footprint only.  One workgroup per batch:
//   Phase 0: async-stage the batch's 2KB of points into LDS (ASYNCcnt DMA)
//   Phase 2: zero a 64KB LDS image
//   Phase 3: scatter with ds_max_u32 atomics (u32 order == f32 order, nonneg)
//   Phase 4: stream the tile LDS -> global with global_store_async_from_lds_b128
// Problem is launch-latency / 1MB-write bound; this is the minimal-work shape.

#define WGRID 128
#define HGRID 128
#define NUMP  256

__global__ __launch_bounds__(256)
void layout_point_kernel(const float* __restrict__ in, float* __restrict__ out)
{
    __shared__ float sm[WGRID * HGRID];          // 64 KB image tile
    const int t = threadIdx.x;                   // 0..255
    const int b = blockIdx.x;                    // batch

    // ---- Phase 0: async-stage this batch's 512 floats (2 KB) into the first 512 floats of LDS
    // wave 0 only: lane l moves 16B per instruction, 4 instructions = 2 KB.
    if (t < 32) {
        unsigned long long g =
            (unsigned long long)(const void*)(in + (size_t)b * (2 * NUMP) + t * 4);
        unsigned l = (unsigned)(unsigned long long)(const void*)(&sm[t * 4]);
        #pragma unroll
        for (int k = 0; k < 4; ++k) {
            asm volatile("global_load_async_to_lds_b128 %0, %1, off"
                         :: "v"(l + (unsigned)(k * 512)),
                            "v"(g + (unsigned long long)(k * 512))
                         : "memory");
        }
        asm volatile("s_wait_asynccnt 0" ::: "memory");
    }
    __syncthreads();

    // ---- Phase 1: each thread owns one point; pull it into registers.
    const float x = 127.0f * sm[t];              // pixel-space center, x-grid
    const float y = 127.0f * sm[NUMP + t];       // pixel-space center, y-grid
    __syncthreads();                             // reads done before we overwrite

    // ---- Phase 2: zero the 64 KB LDS image (16 float4 per thread).
    float4* smv = (float4*)sm;
    const float4 z = make_float4(0.f, 0.f, 0.f, 0.f);
    #pragma unroll
    for (int k = 0; k < 16; ++k)
        smv[k * 256 + t] = z;
    __syncthreads();

    // ---- Phase 3: scatter the 2x2 tent footprint with LDS atomic max.
    // Values are nonnegative, so IEEE f32 ordering == u32 ordering on the bits.
    unsigned* smu = (unsigned*)sm;
    int i0 = (int)x; i0 = (i0 < 0) ? 0 : ((i0 > WGRID - 1) ? WGRID - 1 : i0);
    int j0 = (int)y; j0 = (j0 < 0) ? 0 : ((j0 > HGRID - 1) ? HGRID - 1 : j0);
    const float fx = x - (float)i0, fy = y - (float)j0;
    const float wx0 = 1.0f - fx, wx1 = fx;
    const float wy0 = 1.0f - fy, wy1 = fy;
    const int base = i0 * HGRID + j0;
    atomicMax(&smu[base], __float_as_uint(wx0 * wy0));
    if (j0 + 1 < HGRID)
        atomicMax(&smu[base + 1], __float_as_uint(wx0 * wy1));
    if (i0 + 1 < WGRID)
        atomicMax(&smu[base + HGRID], __float_as_uint(wx1 * wy0));
    if (i0 + 1 < WGRID && j0 + 1 < HGRID)
        atomicMax(&smu[base + HGRID + 1], __float_as_uint(wx1 * wy1));
    __syncthreads();   // all ds_max_u32 complete before async engine reads LDS

    // ---- Phase 4: stream the composited tile LDS -> global via async DMA.
    // Each thread issues 16 x 16B async stores; no VGPR data staging.
    {
        const unsigned long long gbase =
            (unsigned long long)(const void*)(out + (size_t)b * (WGRID * HGRID));
        const unsigned lbase = (unsigned)(unsigned long long)(const void*)sm;
        #pragma unroll
        for (int k = 0; k < 16; ++k) {
            const unsigned off = (unsigned)((k * 256 + t) * 16);
            asm volatile("global_store_async_from_lds_b128 %0, %1, off"
                         :: "v"(gbase + off), "v"(lbase + off) : "memory");
        }
        asm volatile("s_wait_asynccnt 0" ::: "memory");
    }
}

extern "C" void kernel_launch(void* const* d_in, const int* in_sizes, int n_in,
                              void* d_out, int out_size, void* d_ws, size_t ws_size,
                              hipStream_t stream)
{
    const float* in  = (const float*)d_in[0];   // (B, 2, NUM) float32
    float*       out = (float*)d_out;           // (B, 1, 128, 128) float32
    const int B = in_sizes[0] / (2 * NUMP);     // = 16
    layout_point_kernel<<<B, 256, 0, stream>>>(in, out);
}